// GumbelQuantize_13340168421722
// MI455X (gfx1250) — compile-verified
//
#include <hip/hip_runtime.h>
#include <hip/hip_bf16.h>

// ---------------------------------------------------------------------------
// GumbelQuantize for MI455X (gfx1250, wave32)
//  x: [B=32, C=512, H=64, W=64] f32 logits   (channel stride = 16KB)
//  u: [B, H, W, C] f32 uniform noise         (channel contiguous)
//  out = concat( z_q [B,C,H,W], 0.0f, embed_ind [B,H,W] (as float), perplexity )
// Bandwidth-bound (~0.8GB @ 23.3TB/s ≈ 35us). LDS-transpose staging with
// CDNA5 async global->LDS copies; wave32 shuffle reductions.
// ---------------------------------------------------------------------------

#define AS1 __attribute__((address_space(1)))
#define AS3 __attribute__((address_space(3)))

typedef int v2i_t __attribute__((ext_vector_type(2)));

constexpr int       C_      = 512;
constexpr int       HW_     = 4096;                 // 64*64
constexpr int       B_      = 32;
constexpr int       TILE_P  = 32;                   // positions per workgroup
constexpr int       XSTR    = 34;                   // LDS row stride (floats): 8B aligned,
                                                    // 34*lane mod 64 injective -> no bank conflicts
constexpr long long NZ      = (long long)B_ * C_ * HW_;   // 67,108,864 z_q elements
constexpr long long NPOS    = (long long)B_ * HW_;        // 131,072 positions

__global__ void gq_zero_counts(unsigned int* __restrict__ counts) {
  counts[threadIdx.x] = 0u;   // 512 threads, 1 block
}

__global__ __launch_bounds__(256)
void gq_main(const float* __restrict__ x,
             const float* __restrict__ u,
             float* __restrict__ zq,          // d_out base (z_q region)
             float* __restrict__ ind_out,     // d_out + NZ + 1
             unsigned int* __restrict__ counts) {
  __shared__ float lds_x[C_ * XSTR];          // 68KB: x tile, transposed [c][p]

  const int       tid  = threadIdx.x;
  const long long pos0 = (long long)blockIdx.x * TILE_P;   // first global position
  const int       b    = (int)(pos0 / HW_);
  const int       hw0  = (int)(pos0 % HW_);                // multiple of 32
  const float*    xg   = x  + (size_t)b * C_ * HW_ + hw0;  // base of tile, c=0
  float*          zg   = zq + (size_t)b * C_ * HW_ + hw0;

  // -------- Stage A: coalesced x tile -> LDS (async, CDNA5 path) ----------
  // 512 rows x 32 floats; b64 chunks: 8192 total, 32 per thread.
#if __has_builtin(__builtin_amdgcn_global_load_async_to_lds_b64) && \
    __has_builtin(__builtin_amdgcn_s_wait_asynccnt)
  #pragma unroll
  for (int j = 0; j < 32; ++j) {
    const int idx = j * 256 + tid;            // 0..8191
    const int c   = idx >> 4;                 // 0..511
    const int ch  = idx & 15;                 // 16 x 8B chunks per row
    __builtin_amdgcn_global_load_async_to_lds_b64(
        (AS1 v2i_t*)(xg + (size_t)c * HW_ + ch * 2),
        (AS3 v2i_t*)(&lds_x[c * XSTR + ch * 2]),
        0, 0);
  }
  __builtin_amdgcn_s_wait_asynccnt(0);
#else
  #pragma unroll
  for (int j = 0; j < 32; ++j) {
    const int idx = j * 256 + tid;
    const int c   = idx >> 4;
    const int ch  = idx & 15;
    const float2 v = *(const float2*)(xg + (size_t)c * HW_ + ch * 2);
    *(float2*)(&lds_x[c * XSTR + ch * 2]) = v;
  }
#endif
  __syncthreads();

  // -------- Stage B: one wave per position; softmax/argmax/ST -------------
  const int wave = tid >> 5;
  const int lane = tid & 31;

  for (int k = 0; k < 4; ++k) {
    const int       p   = wave * 4 + k;
    const long long pos = pos0 + p;
    const float*    up  = u + (size_t)pos * C_;   // contiguous 2KB -> coalesced

    float s[16];
    float bestv = -3.402823466e+38f;
    int   besti = C_;

    #pragma unroll
    for (int i = 0; i < 16; ++i) {
      const int   c  = lane + 32 * i;
      const float uv = up[c];                                   // 128B/wave
      const float g  = -__logf(-__logf(uv + 1e-20f) + 1e-20f);  // gumbel
      const float sv = lds_x[c * XSTR + p] + g;                 // conflict-free
      s[i] = sv;
      if (sv > bestv) { bestv = sv; besti = c; }                // first occurrence
    }
    // wave32 argmax reduce (smaller index wins ties -> matches jnp.argmax)
    #pragma unroll
    for (int off = 16; off > 0; off >>= 1) {
      const float ov = __shfl_xor(bestv, off, 32);
      const int   oi = __shfl_xor(besti, off, 32);
      if (ov > bestv || (ov == bestv && oi < besti)) { bestv = ov; besti = oi; }
    }
    // softmax (max-shifted) + sum reduce
    float sum = 0.0f;
    #pragma unroll
    for (int i = 0; i < 16; ++i) {
      const float e = __expf(s[i] - bestv);
      s[i] = e;
      sum += e;
    }
    #pragma unroll
    for (int off = 16; off > 0; off >>= 1) sum += __shfl_xor(sum, off, 32);
    const float rs = 1.0f / sum;

    // straight-through: z = (onehot - y) + y ; overwrite wave-private column
    #pragma unroll
    for (int i = 0; i < 16; ++i) {
      const int   c    = lane + 32 * i;
      const float y    = s[i] * rs;
      const float hard = (c == besti) ? 1.0f : 0.0f;
      lds_x[c * XSTR + p] = (hard - y) + y;
    }
    if (lane == 0) {
      ind_out[pos] = (float)besti;
      atomicAdd(&counts[besti], 1u);
    }
  }
  __syncthreads();

  // -------- Stage C: coalesced z_q writeout (128B per wave-row) -----------
  #pragma unroll 4
  for (int j = 0; j < 64; ++j) {
    const int c = j * 8 + (tid >> 5);
    const int p = tid & 31;
    zg[(size_t)c * HW_ + p] = lds_x[c * XSTR + p];   // banks (34c+p)%64 distinct
  }
}

__global__ __launch_bounds__(512)
void gq_finalize(const unsigned int* __restrict__ counts, float* __restrict__ out) {
  __shared__ float red[16];
  const int tid = threadIdx.x;                        // 512 threads = 512 bins
  const float p = (float)counts[tid] * (1.0f / (float)NPOS);
  float t = p * logf(p + 1e-10f);
  #pragma unroll
  for (int off = 16; off > 0; off >>= 1) t += __shfl_xor(t, off, 32);
  if ((tid & 31) == 0) red[tid >> 5] = t;
  __syncthreads();
  if (tid == 0) {
    float s = 0.0f;
    #pragma unroll
    for (int i = 0; i < 16; ++i) s += red[i];
    out[NZ]            = 0.0f;          // "loss" scalar
    out[NZ + 1 + NPOS] = expf(-s);      // perplexity
  }
}

extern "C" void kernel_launch(void* const* d_in, const int* in_sizes, int n_in,
                              void* d_out, int out_size, void* d_ws, size_t ws_size,
                              hipStream_t stream) {
  (void)in_sizes; (void)n_in; (void)out_size; (void)ws_size;
  const float* x = (const float*)d_in[0];
  const float* u = (const float*)d_in[1];
  float*       out    = (float*)d_out;
  unsigned int* counts = (unsigned int*)d_ws;         // 512 * 4B scratch

  gq_zero_counts<<<1, 512, 0, stream>>>(counts);
  gq_main<<<(int)(NPOS / TILE_P), 256, 0, stream>>>(x, u, out, out + NZ + 1, counts);
  gq_finalize<<<1, 512, 0, stream>>>(counts, out);
}